// EdgeNetwork_49082886258998
// MI455X (gfx1250) — compile-verified
//
#include <hip/hip_runtime.h>
#include <math.h>

typedef __attribute__((ext_vector_type(2))) float v2f;
typedef __attribute__((ext_vector_type(4))) float f4;
typedef __attribute__((ext_vector_type(8))) float v8f;

constexpr int BATCH = 16;
constexpr int NPTS  = 1024;   // reduction dim (n)
constexpr int EDG   = 4096;   // E
constexpr int IND   = 8;      // IN_DIM
constexpr int HIDN  = 8;      // HID
constexpr float LN_EPS = 1e-5f;

constexpr int WAVES_PER_BLK = 4;
constexpr int THREADS       = WAVES_PER_BLK * 32;          // 128 (wave32)
constexpr int E_PER_WAVE    = 16;
constexpr int E_PER_BLK     = WAVES_PER_BLK * E_PER_WAVE;  // 64

__device__ __forceinline__ void ln_tanh8(float* h, const float* __restrict__ g,
                                         const float* __restrict__ be) {
    float m = 0.f;
    #pragma unroll
    for (int j = 0; j < HIDN; ++j) m += h[j];
    m *= 0.125f;
    float v = 0.f;
    #pragma unroll
    for (int j = 0; j < HIDN; ++j) { float d = h[j] - m; v += d * d; }
    v *= 0.125f;
    float inv = rsqrtf(v + LN_EPS);
    #pragma unroll
    for (int j = 0; j < HIDN; ++j) h[j] = tanhf((h[j] - m) * inv * g[j] + be[j]);
}

__global__ __launch_bounds__(THREADS)
void edge_net_fused(const float* __restrict__ X,
                    const float* __restrict__ Ri,
                    const float* __restrict__ Ro,
                    const float* __restrict__ W1, const float* __restrict__ b1,
                    const float* __restrict__ g1, const float* __restrict__ be1,
                    const float* __restrict__ W2, const float* __restrict__ b2,
                    const float* __restrict__ g2, const float* __restrict__ be2,
                    const float* __restrict__ W3, const float* __restrict__ b3,
                    const float* __restrict__ g3, const float* __restrict__ be3,
                    const float* __restrict__ W4, const float* __restrict__ b4,
                    float* __restrict__ out)
{
    __shared__ float Xs[NPTS * IND];        // 32 KB: X[b] staged once per block
    __shared__ float feat[E_PER_BLK][17];   // padded stride vs. bank conflicts

    const int tid  = threadIdx.x;
    const int wv   = tid >> 5;
    const int lane = tid & 31;

    const int blocksPerB = EDG / E_PER_BLK; // 64
    const int b  = blockIdx.x / blocksPerB;
    const int e0 = (blockIdx.x % blocksPerB) * E_PER_BLK;

    // ---- cooperative stage of X[b] (1024 x 8 f32) into LDS ----
    {
        const f4* src = reinterpret_cast<const f4*>(X + (size_t)b * NPTS * IND);
        f4*       dst = reinterpret_cast<f4*>(Xs);
        #pragma unroll
        for (int i = tid; i < NPTS * IND / 4; i += THREADS) dst[i] = src[i];
    }
    __syncthreads();

    // ---- WMMA f32 16x16x4 accumulation over n ----
    // A (16x4): lane(0-15)=M, half selects K{0,1}/K{2,3}; VGPR0=K even, VGPR1=K odd
    // B (4x16): lane(0-15)=N col, same K split; C: VGPR r -> M=r+8*half, lane -> N
    const int col  = lane & 15;
    const int half = lane >> 4;
    const int ew   = e0 + wv * E_PER_WAVE;

    const size_t bbase = (size_t)b * NPTS * EDG;
    const float* pro = Ro + bbase + (size_t)(2 * half) * EDG + ew + col;
    const float* pri = Ri + bbase + (size_t)(2 * half) * EDG + ew + col;

    v8f acc_o = {};   // bo tile  (features 0..7)
    v8f acc_i = {};   // bi tile  (features 8..15)

    const int  xcol  = col & 7;
    const bool valid = col < 8;   // pad N=8 -> 16 with zero columns

    #pragma unroll 4
    for (int n = 0; n < NPTS; n += 4) {
        v2f a_o, a_i, bx;
        a_o.x = pro[0];
        a_o.y = pro[EDG];
        a_i.x = pri[0];
        a_i.y = pri[EDG];
        const int xr = (n + 2 * half) * IND + xcol;
        const float x0 = Xs[xr];
        const float x1 = Xs[xr + IND];
        bx.x = valid ? x0 : 0.0f;
        bx.y = valid ? x1 : 0.0f;
        acc_o = __builtin_amdgcn_wmma_f32_16x16x4_f32(
            false, a_o, false, bx, (short)0, acc_o, false, false);
        acc_i = __builtin_amdgcn_wmma_f32_16x16x4_f32(
            false, a_i, false, bx, (short)0, acc_i, false, false);
        pro += (size_t)4 * EDG;
        pri += (size_t)4 * EDG;
    }

    // ---- scatter C tiles into feat[e_local][0..15] = concat(bo, bi) ----
    if (valid) {
        #pragma unroll
        for (int r = 0; r < 8; ++r) {
            const int el = wv * E_PER_WAVE + r + 8 * half;
            feat[el][col]     = acc_o[r];
            feat[el][8 + col] = acc_i[r];
        }
    }
    __syncthreads();

    // ---- fused MLP tail: one thread per edge row ----
    if (tid < E_PER_BLK) {
        float f[2 * IND];
        #pragma unroll
        for (int i = 0; i < 2 * IND; ++i) f[i] = feat[tid][i];

        float h1[HIDN];
        #pragma unroll
        for (int j = 0; j < HIDN; ++j) {
            float s = b1[j];
            #pragma unroll
            for (int i = 0; i < 2 * IND; ++i) s += f[i] * W1[i * HIDN + j];
            h1[j] = s;
        }
        ln_tanh8(h1, g1, be1);

        float h2[HIDN];
        #pragma unroll
        for (int j = 0; j < HIDN; ++j) {
            float s = b2[j];
            #pragma unroll
            for (int i = 0; i < HIDN; ++i) s += h1[i] * W2[i * HIDN + j];
            h2[j] = s;
        }
        ln_tanh8(h2, g2, be2);

        float h3[HIDN];
        #pragma unroll
        for (int j = 0; j < HIDN; ++j) {
            float s = b3[j];
            #pragma unroll
            for (int i = 0; i < HIDN; ++i) s += h2[i] * W3[i * HIDN + j];
            h3[j] = s;
        }
        ln_tanh8(h3, g3, be3);

        float o = b4[0];
        #pragma unroll
        for (int i = 0; i < HIDN; ++i) o += h3[i] * W4[i];

        out[(size_t)b * EDG + e0 + tid] = o;
    }
}

extern "C" void kernel_launch(void* const* d_in, const int* in_sizes, int n_in,
                              void* d_out, int out_size, void* d_ws, size_t ws_size,
                              hipStream_t stream) {
    (void)in_sizes; (void)n_in; (void)out_size; (void)d_ws; (void)ws_size;
    const float* X   = (const float*)d_in[0];
    const float* Ri  = (const float*)d_in[1];
    const float* Ro  = (const float*)d_in[2];
    const float* W1  = (const float*)d_in[3];
    const float* b1  = (const float*)d_in[4];
    const float* g1  = (const float*)d_in[5];
    const float* be1 = (const float*)d_in[6];
    const float* W2  = (const float*)d_in[7];
    const float* b2  = (const float*)d_in[8];
    const float* g2  = (const float*)d_in[9];
    const float* be2 = (const float*)d_in[10];
    const float* W3  = (const float*)d_in[11];
    const float* b3  = (const float*)d_in[12];
    const float* g3  = (const float*)d_in[13];
    const float* be3 = (const float*)d_in[14];
    const float* W4  = (const float*)d_in[15];
    const float* b4  = (const float*)d_in[16];

    dim3 grid(BATCH * (EDG / E_PER_BLK));   // 1024 blocks
    edge_net_fused<<<grid, THREADS, 0, stream>>>(
        X, Ri, Ro, W1, b1, g1, be1, W2, b2, g2, be2,
        W3, b3, g3, be3, W4, b4, (float*)d_out);
}